// _PolicyGATLayer_37555194036831
// MI455X (gfx1250) — compile-verified
//
#include <hip/hip_runtime.h>

#define IN_DIM   128
#define HEADS    4
#define OUT_DIM  32
#define HD       128          // HEADS*OUT_DIM
#define NEG_SLOPE 0.2f
#define GAT_EPS  1e-16f

typedef float v2f __attribute__((ext_vector_type(2)));
typedef float v8f __attribute__((ext_vector_type(8)));

// ---------------------------------------------------------------------------
// 1) h = x @ W^T via V_WMMA_F32_16X16X4_F32.
//    Block = 256 threads = 8 waves; blockIdx.x = 16-row tile, wave = 16-col tile.
//    A (16x4 f32): lane L<16 -> M=L, K=k0,k0+1 ; lane>=16 -> K=k0+2,k0+3
//    B (4x16 f32): lane L<16 -> N=L, K=k0,k0+1 ; lane>=16 -> K=k0+2,k0+3
//    Both are float2 loads at row*128 + k0 + 2*(lane>>4).
// ---------------------------------------------------------------------------
__global__ __launch_bounds__(256)
void gat_gemm_wmma(const float* __restrict__ x, const float* __restrict__ W,
                   float* __restrict__ h) {
  const int lane = threadIdx.x & 31;
  const int wave = threadIdx.x >> 5;        // column tile 0..7
  const int m    = lane & 15;
  const int kh   = lane >> 4;               // 0 or 1
  const long rowBase = (long)blockIdx.x * 16;

  const float* __restrict__ xrow = x + (rowBase + m) * IN_DIM + 2 * kh;
  const float* __restrict__ wrow = W + ((long)(wave * 16 + m)) * IN_DIM + 2 * kh;

  v8f c = {};
#pragma unroll
  for (int k0 = 0; k0 < IN_DIM; k0 += 4) {
    v2f a = *(const v2f*)(xrow + k0);
    v2f b = *(const v2f*)(wrow + k0);
    // 8 args: (neg_a, A, neg_b, B, c_mod, C, reuse_a, reuse_b)
    c = __builtin_amdgcn_wmma_f32_16x16x4_f32(false, a, false, b, (short)0, c,
                                              false, false);
  }
  // C tile: VGPR r -> row = rowBase + r + 8*kh, col = wave*16 + m
  float* __restrict__ hp = h + (rowBase + 8 * kh) * HD + wave * 16 + m;
#pragma unroll
  for (int r = 0; r < 8; ++r) hp[(long)r * HD] = c[r];
}

// Scalar tail for rows not covered by 16-row tiles (unused for N=50000).
__global__ void gat_gemm_tail(const float* __restrict__ x, const float* __restrict__ W,
                              float* __restrict__ h, int rowStart, int nRows) {
  int t = blockIdx.x * blockDim.x + threadIdx.x;   // (row, col)
  if (t >= nRows * HD) return;
  int row = rowStart + t / HD, col = t % HD;
  const float* xr = x + (long)row * IN_DIM;
  const float* wr = W + (long)col * IN_DIM;
  float acc = 0.f;
  for (int k = 0; k < IN_DIM; ++k) acc += xr[k] * wr[k];
  h[(long)row * HD + col] = acc;
}

// ---------------------------------------------------------------------------
// 2) per-(node, head) attention dot products
// ---------------------------------------------------------------------------
__global__ void gat_dots(const float* __restrict__ h,
                         const float* __restrict__ att_src,
                         const float* __restrict__ att_dst,
                         float* __restrict__ a_src, float* __restrict__ a_dst,
                         int n) {
  int t = blockIdx.x * blockDim.x + threadIdx.x;
  if (t >= n * HEADS) return;
  int node = t >> 2, head = t & 3;
  const float* hp = h + (long)node * HD + head * OUT_DIM;
  const float* as = att_src + head * OUT_DIM;
  const float* ad = att_dst + head * OUT_DIM;
  float s = 0.f, d = 0.f;
#pragma unroll 8
  for (int i = 0; i < OUT_DIM; ++i) {
    float v = hp[i];
    s += v * as[i];
    d += v * ad[i];
  }
  a_src[t] = s;
  a_dst[t] = d;
}

// ---------------------------------------------------------------------------
// 3) init: out = bias (broadcast), segment-max keys = 0 (below every real
//    encoded float), segment sums = 0
// ---------------------------------------------------------------------------
__global__ void gat_init(float* __restrict__ out, const float* __restrict__ bias,
                         unsigned* __restrict__ mkey, float* __restrict__ ssum,
                         int n) {
  int t = blockIdx.x * blockDim.x + threadIdx.x;
  if (t >= n * HD) return;
  out[t] = bias[t & (HD - 1)];
  if (t < n * HEADS) { mkey[t] = 0u; ssum[t] = 0.f; }
}

// Monotone order-preserving float->uint encoding (handles negatives).
__device__ __forceinline__ unsigned enc_f32(float f) {
  unsigned u = __float_as_uint(f);
  return (u & 0x80000000u) ? ~u : (u | 0x80000000u);
}
__device__ __forceinline__ float dec_f32(unsigned k) {
  return (k & 0x80000000u) ? __uint_as_float(k ^ 0x80000000u)
                           : __uint_as_float(~k);
}
__device__ __forceinline__ float leaky(float a) {
  return a > 0.f ? a : NEG_SLOPE * a;
}

// ---------------------------------------------------------------------------
// 4) segment max over edges (per dst, per head) via atomicMax on encoded uint
// ---------------------------------------------------------------------------
__global__ void gat_edge_max(const long long* __restrict__ ei,
                             const float* __restrict__ a_src,
                             const float* __restrict__ a_dst,
                             unsigned* __restrict__ mkey, int e) {
  int t = blockIdx.x * blockDim.x + threadIdx.x;
  if (t >= e * HEADS) return;
  int edge = t >> 2, head = t & 3;
  int src = (int)ei[edge];
  int dst = (int)ei[e + edge];
  float alpha = leaky(a_src[src * HEADS + head] + a_dst[dst * HEADS + head]);
  atomicMax(&mkey[dst * HEADS + head], enc_f32(alpha));
}

// ---------------------------------------------------------------------------
// 5) ex = exp(alpha - m[dst]); store ex, segment-sum via atomicAdd
// ---------------------------------------------------------------------------
__global__ void gat_edge_expsum(const long long* __restrict__ ei,
                                const float* __restrict__ a_src,
                                const float* __restrict__ a_dst,
                                const unsigned* __restrict__ mkey,
                                float* __restrict__ exbuf,
                                float* __restrict__ ssum, int e) {
  int t = blockIdx.x * blockDim.x + threadIdx.x;
  if (t >= e * HEADS) return;
  int edge = t >> 2, head = t & 3;
  int src = (int)ei[edge];
  int dst = (int)ei[e + edge];
  float alpha = leaky(a_src[src * HEADS + head] + a_dst[dst * HEADS + head]);
  float ex = __expf(alpha - dec_f32(mkey[dst * HEADS + head]));
  exbuf[t] = ex;
  atomicAdd(&ssum[dst * HEADS + head], ex);
}

// ---------------------------------------------------------------------------
// 6) aggregate: out[dst, c] += h[src, c] * alpha[e, head(c)]
//    One thread per (edge, channel): coalesced h reads, native f32 atomics.
// ---------------------------------------------------------------------------
__global__ void gat_aggregate(const long long* __restrict__ ei,
                              const float* __restrict__ h,
                              const float* __restrict__ exbuf,
                              const float* __restrict__ ssum,
                              float* __restrict__ out, int e) {
  long t = (long)blockIdx.x * blockDim.x + threadIdx.x;
  if (t >= (long)e * HD) return;
  int edge = (int)(t >> 7);
  int c    = (int)(t & (HD - 1));
  int head = c >> 5;
  int src = (int)ei[edge];
  int dst = (int)ei[e + edge];
  float ex = exbuf[edge * HEADS + head];
  float al = ex / (ssum[dst * HEADS + head] + GAT_EPS);
  atomicAdd(&out[(long)dst * HD + c], h[(long)src * HD + c] * al);
}

// ---------------------------------------------------------------------------
// 7) alpha_pooled[e] = mean over heads of normalized alpha
// ---------------------------------------------------------------------------
__global__ void gat_pooled(const long long* __restrict__ ei,
                           const float* __restrict__ exbuf,
                           const float* __restrict__ ssum,
                           float* __restrict__ pooled, int e) {
  int edge = blockIdx.x * blockDim.x + threadIdx.x;
  if (edge >= e) return;
  int dst = (int)ei[e + edge];
  float acc = 0.f;
#pragma unroll
  for (int head = 0; head < HEADS; ++head) {
    acc += exbuf[edge * HEADS + head] /
           (ssum[dst * HEADS + head] + GAT_EPS);
  }
  pooled[edge] = acc * (1.0f / HEADS);
}

// ---------------------------------------------------------------------------
// launcher
// ---------------------------------------------------------------------------
extern "C" void kernel_launch(void* const* d_in, const int* in_sizes, int n_in,
                              void* d_out, int out_size, void* d_ws, size_t ws_size,
                              hipStream_t stream) {
  (void)n_in; (void)out_size; (void)ws_size;
  const float*     x       = (const float*)d_in[0];
  const long long* ei      = (const long long*)d_in[1];
  const float*     W       = (const float*)d_in[2];
  const float*     att_src = (const float*)d_in[3];
  const float*     att_dst = (const float*)d_in[4];
  const float*     bias    = (const float*)d_in[5];

  const int N = in_sizes[0] / IN_DIM;
  const int E = in_sizes[1] / 2;

  float* out    = (float*)d_out;             // N*HD floats
  float* pooled = out + (size_t)N * HD;      // E floats

  char* ws = (char*)d_ws;
  float*    h     = (float*)ws;     ws += sizeof(float)    * (size_t)N * HD;
  float*    a_src = (float*)ws;     ws += sizeof(float)    * (size_t)N * HEADS;
  float*    a_dst = (float*)ws;     ws += sizeof(float)    * (size_t)N * HEADS;
  unsigned* mkey  = (unsigned*)ws;  ws += sizeof(unsigned) * (size_t)N * HEADS;
  float*    ssum  = (float*)ws;     ws += sizeof(float)    * (size_t)N * HEADS;
  float*    exbuf = (float*)ws;     ws += sizeof(float)    * (size_t)E * HEADS;

  const int B = 256;

  // 1) GEMM (WMMA path covers N - N%16 rows; tail kernel for the rest)
  int fullTiles = N / 16;
  if (fullTiles > 0)
    gat_gemm_wmma<<<fullTiles, B, 0, stream>>>(x, W, h);
  int remRows = N - fullTiles * 16;
  if (remRows > 0)
    gat_gemm_tail<<<(remRows * HD + B - 1) / B, B, 0, stream>>>(
        x, W, h, fullTiles * 16, remRows);

  // 2) attention dots
  gat_dots<<<(N * HEADS + B - 1) / B, B, 0, stream>>>(h, att_src, att_dst,
                                                      a_src, a_dst, N);

  // 3) init out/bias + segment accumulators
  gat_init<<<(N * HD + B - 1) / B, B, 0, stream>>>(out, bias, mkey, ssum, N);

  // 4) segment max
  gat_edge_max<<<(E * HEADS + B - 1) / B, B, 0, stream>>>(ei, a_src, a_dst,
                                                          mkey, E);

  // 5) exp + segment sum
  gat_edge_expsum<<<(E * HEADS + B - 1) / B, B, 0, stream>>>(
      ei, a_src, a_dst, mkey, exbuf, ssum, E);

  // 6) weighted scatter-add aggregation
  long aggThreads = (long)E * HD;
  gat_aggregate<<<(unsigned)((aggThreads + B - 1) / B), B, 0, stream>>>(
      ei, h, exbuf, ssum, out, E);

  // 7) pooled alpha
  gat_pooled<<<(E + B - 1) / B, B, 0, stream>>>(ei, exbuf, ssum, pooled, E);
}